// RNN_18528488915578
// MI455X (gfx1250) — compile-verified
//
#include <hip/hip_runtime.h>
#include <math.h>

// ---------------------------------------------------------------------------
// MI455X (gfx1250) GRU step + 50257-wide projection + log_softmax.
//
// All heavy work is fp32 GEMV: 1.63 GB of weights read exactly once ->
// HBM-bound, ~70us floor at 23.3 TB/s. Inner loop: V_WMMA_F32_16X16X4_F32,
// 16 rows per wave. To cover the BW*latency product (~11.6 MB in flight):
//   - UF=16 batched b64 loads per wave (4 KB outstanding per wave)
//   - K split in 2 over blockIdx.z (bias-free partials, summed in epilogues)
// ---------------------------------------------------------------------------

typedef __attribute__((ext_vector_type(2))) float v2f;
typedef __attribute__((ext_vector_type(4))) float v4f;
typedef __attribute__((ext_vector_type(8))) float v8f;

#if defined(__has_builtin)
#if __has_builtin(__builtin_amdgcn_wmma_f32_16x16x4_f32)
#define USE_WMMA_F32 1
#endif
#endif

#define KD 4096            // I == H : K dim of every GEMV
#define KSPLIT 2
#define KCHUNK (KD / KSPLIT)
#define HN 4096
#define ON 50257
#define UF 16              // outstanding b64 loads per wave per batch

struct GemvArgs {
  const float* W;   // [nRows, KD] row-major
  const float* x;   // [KD]
  float*       y0;  // partial sums, z = 0
  float*       y1;  // partial sums, z = 1
  int nRows;
  int reluX;        // apply relu while staging x (fuses relu(h2))
};

// Block = 256 threads = 8 waves; wave w -> 16 rows; blockIdx.z -> K half.
__global__ __launch_bounds__(256) void gemv_wmma_kernel(GemvArgs a0, GemvArgs a1) {
  GemvArgs A = (blockIdx.y == 0) ? a0 : a1;

  __shared__ float xs[KD];
  {
    const v4f* xin = (const v4f*)A.x;
    v4f* xsh = (v4f*)xs;
    const bool relu = (A.reluX != 0);
    for (int i = threadIdx.x; i < KD / 4; i += 256) {
      v4f v = xin[i];
      if (relu) {
        v.x = fmaxf(v.x, 0.f); v.y = fmaxf(v.y, 0.f);
        v.z = fmaxf(v.z, 0.f); v.w = fmaxf(v.w, 0.f);
      }
      xsh[i] = v;
    }
  }
  __syncthreads();

  const int wave = threadIdx.x >> 5;
  const int lane = threadIdx.x & 31;
  const int hi   = lane >> 4;       // lane half selects K sub-pair (ISA 7.12.2)
  const int lo   = lane & 15;       // A-matrix: M = lane % 16
  const int kz   = blockIdx.z;      // K half

  const int rowBase = (blockIdx.x * 8 + wave) * 16;
  if (rowBase >= A.nRows) return;   // after the only barrier: safe

  int row = rowBase + lo;
  if (row > A.nRows - 1) row = A.nRows - 1;  // clamp OOB loads; stores guarded

  const v2f* Wr2 = (const v2f*)(A.W + (size_t)row * KD);
  const v2f* xs2 = (const v2f*)xs;
  float* y = kz ? A.y1 : A.y0;

  const int k2Base = kz * (KCHUNK / 2);      // v2f units
  const int k2End  = k2Base + (KCHUNK / 2);

#if USE_WMMA_F32
  v8f c = {};
  for (int k2 = k2Base + hi; k2 < k2End; k2 += 2 * UF) {
    v2f a[UF], bb[UF];
#pragma unroll
    for (int u = 0; u < UF; ++u) a[u] = Wr2[k2 + 2 * u];     // 16 b64 in flight
#pragma unroll
    for (int u = 0; u < UF; ++u) bb[u] = xs2[k2 + 2 * u];    // B replicated over N
#pragma unroll
    for (int u = 0; u < UF; ++u)
      c = __builtin_amdgcn_wmma_f32_16x16x4_f32(
          /*neg_a=*/false, a[u], /*neg_b=*/false, bb[u],
          /*c_mod=*/(short)0, c, /*reuse_a=*/false, /*reuse_b=*/false);
  }
  // D layout: VGPR j -> M=j (lanes 0-15) / M=j+8 (lanes 16-31); N = lane%16.
  // All N columns identical; lanes with N==0 write rows.
  if (lo == 0) {
    const int r0 = rowBase + 8 * hi;
#pragma unroll
    for (int j = 0; j < 8; ++j) {
      const int r = r0 + j;
      if (r < A.nRows) y[r] = c[j];
    }
  }
#else
  float p = 0.f;
  for (int k2 = k2Base + hi; k2 < k2End; k2 += 2 * UF) {
    v2f a[UF], bb[UF];
#pragma unroll
    for (int u = 0; u < UF; ++u) a[u] = Wr2[k2 + 2 * u];
#pragma unroll
    for (int u = 0; u < UF; ++u) bb[u] = xs2[k2 + 2 * u];
#pragma unroll
    for (int u = 0; u < UF; ++u) { p = fmaf(a[u].x, bb[u].x, p); p = fmaf(a[u].y, bb[u].y, p); }
  }
  p += __shfl_xor(p, 16, 32);
  if (hi == 0) {
    const int r = rowBase + lo;
    if (r < A.nRows) y[r] = p;
  }
#endif
}

// Gate preactivations = (gi partials z0+z1) + b_ih + (gh partials) + b_hh.
// PyTorch gate order r,z,n. h' = (1-z)*n + z*h_prev.
__global__ void gru_combine_kernel(const float* __restrict__ gi,   // [2][3H]
                                   const float* __restrict__ gh,   // [2][3H]
                                   const float* __restrict__ b_ih,
                                   const float* __restrict__ b_hh,
                                   const float* __restrict__ hprev,
                                   float* __restrict__ hnew, int Hn) {
  int m = blockIdx.x * blockDim.x + threadIdx.x;
  if (m >= Hn) return;
  const int T = 3 * Hn;
  float ir = gi[m]            + gi[T + m]            + b_ih[m];
  float iz = gi[Hn + m]       + gi[T + Hn + m]       + b_ih[Hn + m];
  float in = gi[2 * Hn + m]   + gi[T + 2 * Hn + m]   + b_ih[2 * Hn + m];
  float hr = gh[m]            + gh[T + m]            + b_hh[m];
  float hz = gh[Hn + m]       + gh[T + Hn + m]       + b_hh[Hn + m];
  float hn = gh[2 * Hn + m]   + gh[T + 2 * Hn + m]   + b_hh[2 * Hn + m];
  float r = 1.f / (1.f + __expf(-(ir + hr)));
  float z = 1.f / (1.f + __expf(-(iz + hz)));
  float n = tanhf(in + r * hn);
  hnew[m] = (1.f - z) * n + z * hprev[m];
}

// ---- log_softmax over l(i) = p0[i] + p1[i] + b[i], 3-stage deterministic ----
#define SB 128   // reduction blocks

__global__ __launch_bounds__(256) void lse_pass1(const float* __restrict__ p0,
                                                 const float* __restrict__ p1,
                                                 const float* __restrict__ bias,
                                                 float* __restrict__ pmax,
                                                 float* __restrict__ psum, int N) {
  __shared__ float smax[256], ssum[256];
  const int t = threadIdx.x;
  const int stride = gridDim.x * 256;
  float m = -3.402823466e38f, s = 0.f;
  for (int i = blockIdx.x * 256 + t; i < N; i += stride) {
    float v = p0[i] + p1[i] + bias[i];
    float nm = fmaxf(m, v);
    s = s * __expf(m - nm) + __expf(v - nm);   // online softmax merge
    m = nm;
  }
  smax[t] = m; ssum[t] = s; __syncthreads();
  for (int q = 128; q > 0; q >>= 1) {
    if (t < q) {
      float m2 = smax[t + q], s2 = ssum[t + q];
      float nm = fmaxf(smax[t], m2);
      ssum[t] = ssum[t] * __expf(smax[t] - nm) + s2 * __expf(m2 - nm);
      smax[t] = nm;
    }
    __syncthreads();
  }
  if (t == 0) { pmax[blockIdx.x] = smax[0]; psum[blockIdx.x] = ssum[0]; }
}

__global__ __launch_bounds__(SB) void lse_pass2(const float* __restrict__ pmax,
                                                const float* __restrict__ psum,
                                                float* __restrict__ lse) {
  __shared__ float smax[SB], ssum[SB];
  const int t = threadIdx.x;
  smax[t] = pmax[t]; ssum[t] = psum[t]; __syncthreads();
  for (int q = SB / 2; q > 0; q >>= 1) {
    if (t < q) {
      float m2 = smax[t + q], s2 = ssum[t + q];
      float nm = fmaxf(smax[t], m2);
      ssum[t] = ssum[t] * __expf(smax[t] - nm) + s2 * __expf(m2 - nm);
      smax[t] = nm;
    }
    __syncthreads();
  }
  if (t == 0) *lse = smax[0] + __logf(ssum[0]);
}

__global__ __launch_bounds__(256) void lse_pass3(const float* __restrict__ p0,
                                                 const float* __restrict__ p1,  // == out (overwritten)
                                                 const float* __restrict__ bias,
                                                 const float* __restrict__ lse,
                                                 float* __restrict__ out, int N) {
  const float L = *lse;
  const int stride = gridDim.x * 256;
  for (int i = blockIdx.x * 256 + threadIdx.x; i < N; i += stride)
    out[i] = p0[i] + p1[i] + bias[i] - L;
}

extern "C" void kernel_launch(void* const* d_in, const int* in_sizes, int n_in,
                              void* d_out, int out_size, void* d_ws, size_t ws_size,
                              hipStream_t stream) {
  const float* input  = (const float*)d_in[0];
  const float* hidden = (const float*)d_in[1];
  const float* w_ih1  = (const float*)d_in[2];
  const float* w_hh1  = (const float*)d_in[3];
  const float* b_ih1  = (const float*)d_in[4];
  const float* b_hh1  = (const float*)d_in[5];
  const float* w_ih2  = (const float*)d_in[6];
  const float* w_hh2  = (const float*)d_in[7];
  const float* b_ih2  = (const float*)d_in[8];
  const float* b_hh2  = (const float*)d_in[9];
  const float* w_out  = (const float*)d_in[10];
  const float* b_out  = (const float*)d_in[11];
  float* out = (float*)d_out;               // [ON logits][HN h2]

  float* ws   = (float*)d_ws;
  float* gi1  = ws;                         // [2][3H]
  float* gh1  = ws + 6 * HN;                // [2][3H]
  float* gi2  = ws + 12 * HN;               // [2][3H]
  float* gh2  = ws + 18 * HN;               // [2][3H]
  float* h1   = ws + 24 * HN;               // [H]
  float* lp0  = ws + 25 * HN;               // [ON] logits partial z=0
  float* pmax = lp0 + ON;                   // [SB]
  float* psum = pmax + SB;                  // [SB]
  float* lse  = psum + SB;                  // [1]
  float* h2   = out + ON;                   // second reference output
  float* lp1  = out;                        // logits partial z=1 parked in out[]

  const dim3 blk(256);
  const dim3 gridGru((3 * HN) / 128, 2, KSPLIT);      // 96 x 2 x 2
  const dim3 gridOut((ON + 127) / 128, 1, KSPLIT);    // 393 x 1 x 2

  // ---- GRU layer 1: ih & hh GEMVs concurrent, K split in 2 ----
  {
    GemvArgs a0{w_ih1, input,  gi1, gi1 + 3 * HN, 3 * HN, 0};
    GemvArgs a1{w_hh1, hidden, gh1, gh1 + 3 * HN, 3 * HN, 0};
    gemv_wmma_kernel<<<gridGru, blk, 0, stream>>>(a0, a1);
  }
  gru_combine_kernel<<<HN / 256, blk, 0, stream>>>(gi1, gh1, b_ih1, b_hh1, hidden, h1, HN);

  // ---- GRU layer 2 (x == h == h1) ----
  {
    GemvArgs a0{w_ih2, h1, gi2, gi2 + 3 * HN, 3 * HN, 0};
    GemvArgs a1{w_hh2, h1, gh2, gh2 + 3 * HN, 3 * HN, 0};
    gemv_wmma_kernel<<<gridGru, blk, 0, stream>>>(a0, a1);
  }
  gru_combine_kernel<<<HN / 256, blk, 0, stream>>>(gi2, gh2, b_ih2, b_hh2, h1, h2, HN);

  // ---- logits partials = W_out relu(h2), K split in 2 ----
  {
    GemvArgs a0{w_out, h2, lp0, lp1, ON, 1};
    gemv_wmma_kernel<<<gridOut, blk, 0, stream>>>(a0, a0);
  }

  // ---- log_softmax: online (max,sum) tree, then write out = l - lse ----
  lse_pass1<<<SB, blk, 0, stream>>>(lp0, lp1, b_out, pmax, psum, ON);
  lse_pass2<<<1, SB, 0, stream>>>(pmax, psum, lse);
  lse_pass3<<<SB, blk, 0, stream>>>(lp0, lp1, b_out, lse, out, ON);
}